// _DiagS4DLayer_84361747628485
// MI455X (gfx1250) — compile-verified
//
#include <hip/hip_runtime.h>

#define H_DIM 256
#define N_DIM 64
#define B_DIM 4
#define L_DIM 2048

typedef __attribute__((ext_vector_type(2))) float v2f;
typedef __attribute__((ext_vector_type(8))) float v8f;

// ---------------------------------------------------------------------------
// Kernel 1: transpose u (B,L,H) -> uT (B,H,L), LDS 32x32 tiles, coalesced.
// ---------------------------------------------------------------------------
__global__ void __launch_bounds__(256)
k_transpose_u(const float* __restrict__ u, float* __restrict__ uT) {
    __shared__ float tile[32][33];
    const int nLt = L_DIM / 32;            // 64
    const int nHt = H_DIM / 32;            // 8
    int bid = blockIdx.x;
    int b   = bid / (nLt * nHt);
    int rem = bid % (nLt * nHt);
    int lt  = rem / nHt;
    int ht  = rem % nHt;
    int tx  = threadIdx.x & 31;
    int ty  = threadIdx.x >> 5;            // 0..7
    const int l0 = lt * 32, h0 = ht * 32;
    #pragma unroll
    for (int r = ty; r < 32; r += 8)
        tile[r][tx] = u[((size_t)b * L_DIM + (l0 + r)) * H_DIM + h0 + tx];
    __syncthreads();
    #pragma unroll
    for (int r = ty; r < 32; r += 8)
        uT[((size_t)b * H_DIM + (h0 + r)) * L_DIM + l0 + tx] = tile[tx][r];
}

// ---------------------------------------------------------------------------
// Kernel 2: K[h,l] = sum_n Re(CB * exp(s*l)); per-h derived params in LDS.
// ---------------------------------------------------------------------------
__global__ void __launch_bounds__(256)
k_kernelgen(const float* __restrict__ log_neg_real, const float* __restrict__ imag,
            const float* __restrict__ B_re, const float* __restrict__ B_im,
            const float* __restrict__ C_re, const float* __restrict__ C_im,
            const float* __restrict__ log_step, float* __restrict__ Kg) {
    const int h = blockIdx.x;
    const int tid = threadIdx.x;
    __shared__ float CBre[N_DIM], CBim[N_DIM], sre[N_DIM], sim[N_DIM];
    if (tid < N_DIM) {
        float step = __expf(log_step[h]);
        step = fminf(fmaxf(step, 1e-5f), 1.0f);
        float lam = -__expf(log_neg_real[tid]);
        sre[tid] = step * lam;
        sim[tid] = step * imag[tid];
        float bre = B_re[h * N_DIM + tid] * step;
        float bim = B_im[h * N_DIM + tid] * step;
        float cre = C_re[h * N_DIM + tid];
        float cim = C_im[h * N_DIM + tid];
        CBre[tid] = cre * bre - cim * bim;
        CBim[tid] = cre * bim + cim * bre;
    }
    __syncthreads();
    for (int l = tid; l < L_DIM; l += 256) {
        const float fl = (float)l;
        float acc = 0.0f;
        #pragma unroll 4
        for (int n = 0; n < N_DIM; ++n) {
            float e = __expf(sre[n] * fl);
            float sn, cs;
            __sincosf(sim[n] * fl, &sn, &cs);
            acc += e * (CBre[n] * cs - CBim[n] * sn);
        }
        Kg[(size_t)h * L_DIM + l] = acc;
    }
}

// ---------------------------------------------------------------------------
// Kernel 3: causal conv via block-Toeplitz WMMA (V_WMMA_F32_16X16X4_F32).
//   grid = (L/256 groups, H heads); block = 128 threads = 4 waves = 4 batches.
//   Wave computes D[t, i] = y[h,b, g*256 + 16*i + t] for t,i in 0..15:
//     D = sum_{d=0}^{g*16+15}  A(d) * B(d)
//     A(d)[t,k] = K[h, 16d + t - k]              (zero-pad -> causality)
//     B(d)[k,i] = uT[b,h, (g*16 + i - d)*16 + k] (zero-pad left edge)
//   fused epilogue: y += D[h]*u, stored to yT (B,H,L).
// ---------------------------------------------------------------------------
__global__ void __launch_bounds__(128)
k_conv_wmma(const float* __restrict__ Kg, const float* __restrict__ uT,
            const float* __restrict__ Dv, float* __restrict__ yT) {
    const int h    = blockIdx.y;
    const int g    = blockIdx.x;      // 0..7 (group of 16 output blocks)
    const int g16  = g * 16;
    const int tid  = threadIdx.x;
    const int wave = tid >> 5;        // batch index b
    const int lane = tid & 31;

    __shared__ alignas(16) float Kpad[16 + L_DIM];          // 16 leading zeros
    __shared__ alignas(16) float upad[B_DIM][240 + L_DIM];  // 240 leading zeros

    const int qcnt = g16 * 16 + 256;  // prefix length needed (multiple of 256)

    const float* Krow = Kg + (size_t)h * L_DIM;
    const float* urow = uT + ((size_t)wave * H_DIM + h) * L_DIM;
    __builtin_prefetch(Krow, 0, 3);            // global_prefetch_b8 (L2 warm)
    __builtin_prefetch(urow, 0, 3);

    // Staging in 16B units: global_load_b128 -> ds_store_b128.
    for (int i = tid; i < 4; i += 128)
        ((float4*)Kpad)[i] = make_float4(0.f, 0.f, 0.f, 0.f);
    {
        const float4* src = (const float4*)Krow;
        float4*       dst = (float4*)(Kpad + 16);
        for (int i = tid; i < (qcnt >> 2); i += 128) dst[i] = src[i];
    }
    {
        const float4* src = (const float4*)urow;
        float4*       dst = (float4*)upad[wave];
        for (int i = lane; i < 60; i += 32) dst[i] = make_float4(0.f, 0.f, 0.f, 0.f);
        dst += 60;                                     // skip 240-float pad
        for (int i = lane; i < (qcnt >> 2); i += 32) dst[i] = src[i];
    }
    __syncthreads();

    // ISA lane layouts (wave32):
    //  A 16x4 f32: lane -> M = lane&15 ; VGPR0/1 hold K = 2*(lane>>4) + {0,1}
    //  B 4x16 f32: lane -> N = lane&15 ; VGPR0/1 hold K = 2*(lane>>4) + {0,1}
    //  C/D 16x16:  lane -> N = lane&15 ; VGPR r holds M = r + 8*(lane>>4)
    const int nl    = lane & 15;
    const int khalf = lane >> 4;
    const float* up = &upad[wave][240];

    v8f acc = {};
    const int dmax = g16 + 15;
    for (int d = 0; d <= dmax; ++d) {
        const int aBase = 16 + 16 * d + nl;          // Kpad index for k=0
        const int uBase = (g16 + nl - d) * 16;       // u offset of block (i-d)
        #pragma unroll
        for (int s = 0; s < 4; ++s) {
            const int k = 4 * s + 2 * khalf;
            v2f a;
            a.x = Kpad[aBase - k];                   // adjacent dwords ->
            a.y = Kpad[aBase - k - 1];               //   ds_load_2addr_b32
            // (uBase + k) is even and upad is 16B-aligned -> ds_load_b64
            const v2f bm = *(const v2f*)&up[uBase + k];
            acc = __builtin_amdgcn_wmma_f32_16x16x4_f32(
                /*neg_a=*/false, a, /*neg_b=*/false, bm,
                /*c_mod=*/(short)0, acc, /*reuse_a=*/false, /*reuse_b=*/false);
        }
    }

    // Epilogue: + D[h]*u, two b128 stores per lane (qb is a multiple of 8).
    const float Dh = Dv[h];
    float* yrow = yT + ((size_t)wave * H_DIM + h) * L_DIM;
    const int qb = g16 * 16 + 16 * nl + 8 * khalf;   // M = r + 8*khalf
    const float4 u0 = *(const float4*)&up[qb];
    const float4 u1 = *(const float4*)&up[qb + 4];
    float4 y0, y1;
    y0.x = acc[0] + Dh * u0.x;  y0.y = acc[1] + Dh * u0.y;
    y0.z = acc[2] + Dh * u0.z;  y0.w = acc[3] + Dh * u0.w;
    y1.x = acc[4] + Dh * u1.x;  y1.y = acc[5] + Dh * u1.y;
    y1.z = acc[6] + Dh * u1.z;  y1.w = acc[7] + Dh * u1.w;
    *(float4*)&yrow[qb]     = y0;
    *(float4*)&yrow[qb + 4] = y1;
}

// ---------------------------------------------------------------------------
// Kernel 4: exact GELU + LayerNorm over H, output (B,L,H).
//   block = 256 threads (one per h), handles one b and 16 l values.
// ---------------------------------------------------------------------------
__global__ void __launch_bounds__(256)
k_gelu_ln(const float* __restrict__ yT, const float* __restrict__ gamma,
          const float* __restrict__ beta, float* __restrict__ out) {
    const int b  = blockIdx.y;
    const int l0 = blockIdx.x * 16;
    const int t  = threadIdx.x;                  // h index
    __shared__ float ysh[H_DIM][17];
    __shared__ float rs[256], rq[256];
    __shared__ float mu[16], rsig[16];

    const float* row = yT + ((size_t)b * H_DIM + t) * L_DIM + l0;
    #pragma unroll
    for (int j = 0; j < 16; ++j) {
        float x = row[j];
        ysh[t][j] = 0.5f * x * (1.0f + erff(x * 0.70710678118654752f));
    }
    __syncthreads();

    const int l = t & 15, part = t >> 4;
    float s = 0.0f, q = 0.0f;
    #pragma unroll
    for (int i = 0; i < 16; ++i) {
        float v = ysh[part * 16 + i][l];
        s += v; q += v * v;
    }
    rs[t] = s; rq[t] = q;
    __syncthreads();
    if (t < 16) {
        float ss = 0.0f, qq = 0.0f;
        #pragma unroll
        for (int p = 0; p < 16; ++p) { ss += rs[p * 16 + t]; qq += rq[p * 16 + t]; }
        float m = ss * (1.0f / 256.0f);
        float v = qq * (1.0f / 256.0f) - m * m;
        mu[t]   = m;
        rsig[t] = rsqrtf(v + 1e-5f);
    }
    __syncthreads();
    const float gt = gamma[t], bt = beta[t];
    #pragma unroll
    for (int j = 0; j < 16; ++j)
        out[((size_t)b * L_DIM + (l0 + j)) * H_DIM + t] =
            (ysh[t][j] - mu[j]) * rsig[j] * gt + bt;
}

// ---------------------------------------------------------------------------
extern "C" void kernel_launch(void* const* d_in, const int* in_sizes, int n_in,
                              void* d_out, int out_size, void* d_ws, size_t ws_size,
                              hipStream_t stream) {
    (void)in_sizes; (void)n_in; (void)out_size; (void)ws_size;
    const float* u            = (const float*)d_in[0];
    const float* log_neg_real = (const float*)d_in[1];
    const float* imag         = (const float*)d_in[2];
    const float* B_re         = (const float*)d_in[3];
    const float* B_im         = (const float*)d_in[4];
    const float* C_re         = (const float*)d_in[5];
    const float* C_im         = (const float*)d_in[6];
    const float* Dv           = (const float*)d_in[7];
    const float* log_step     = (const float*)d_in[8];
    const float* ln_gamma     = (const float*)d_in[9];
    const float* ln_beta      = (const float*)d_in[10];
    float* out = (float*)d_out;

    float* ws  = (float*)d_ws;
    float* Kg  = ws;                                        // H*L   floats
    float* uT  = Kg + (size_t)H_DIM * L_DIM;                // B*H*L floats
    float* yT  = uT + (size_t)B_DIM * H_DIM * L_DIM;        // B*H*L floats

    k_transpose_u<<<dim3(B_DIM * (L_DIM / 32) * (H_DIM / 32)), 256, 0, stream>>>(u, uT);
    k_kernelgen<<<dim3(H_DIM), 256, 0, stream>>>(log_neg_real, imag, B_re, B_im,
                                                 C_re, C_im, log_step, Kg);
    k_conv_wmma<<<dim3(L_DIM / 256, H_DIM), 128, 0, stream>>>(Kg, uT, Dv, yT);
    k_gelu_ln<<<dim3(L_DIM / 16, B_DIM), 256, 0, stream>>>(yT, ln_gamma, ln_beta, out);
}